// ACTChunkingRNNLayer_68341519614201
// MI455X (gfx1250) — compile-verified
//
#include <hip/hip_runtime.h>
#include <math.h>

// ACT chunking RNN for MI455X (gfx1250), f32 WMMA path.
// B=64, T=256, D=H=1024.
//   Phase 0: pair-swizzle Wx and Wh:  Wp[p][n] = {W[2p][n], W[2p+1][n]}
//            -> every WMMA B-fragment is ONE global_load_b64.
//   Phase 1: XWx[t,b,:] = features[b,t,:] @ Wx + b      (one WMMA GEMM, 34 GFLOP)
//            mask[t,b]  = (sum |features[b,t,:]| != 0)
//   Phase 2: ONE persistent kernel runs the whole 256-step scan.
//            Grid = 4 blocks (always co-resident) synchronized per step by a
//            device-scope atomic grid barrier; state chains through the
//            inter_states half of d_out. No per-step launch overhead; Whp
//            (4 MB) stays resident in the 192 MB L2 for the entire scan.

typedef __attribute__((ext_vector_type(2))) float v2f;
typedef __attribute__((ext_vector_type(4))) float v4f;
typedef __attribute__((ext_vector_type(8))) float v8f;

#define B_ 64
#define T_ 256
#define D_ 1024
#define H_ 1024
#define NBLK_ 4          // persistent-kernel blocks (16 batch rows each)

// ---------------------------------------------------------------------------
// Pair-swizzle a (H,H) weight matrix: Wp[(p*H + n)*2 + j] = W[(2p+j)*H + n].
// ---------------------------------------------------------------------------
__global__ __launch_bounds__(256) void swizzle_kernel(const float* __restrict__ W,
                                                      float* __restrict__ Wp) {
    int idx = blockIdx.x * blockDim.x + threadIdx.x;   // over (H/2)*H pairs
    if (idx >= (H_ / 2) * H_) return;
    int n = idx % H_;
    int p = idx / H_;
    v2f v;
    v.x = W[(long)(2 * p)     * H_ + n];
    v.y = W[(long)(2 * p + 1) * H_ + n];
    *(v2f*)(Wp + (long)idx * 2) = v;
}

// ---------------------------------------------------------------------------
// fmask[t*B + b] = (sum_d |F[b,t,d]| != 0) ? 1 : 0 ; also zero barrier counter.
// ---------------------------------------------------------------------------
__global__ __launch_bounds__(256) void mask_kernel(const float* __restrict__ F,
                                                   float* __restrict__ mask,
                                                   unsigned* __restrict__ counter) {
    int idx = blockIdx.x * blockDim.x + threadIdx.x;
    if (idx == 0) *counter = 0u;                 // reset grid barrier each call
    if (idx >= B_ * T_) return;
    int b = idx / T_, t = idx % T_;
    const float* row = F + ((long)b * T_ + t) * D_;
    float s = 0.f;
    for (int d = 0; d < D_; d += 4) {
        v4f v = *(const v4f*)(row + d);
        s += fabsf(v.x) + fabsf(v.y) + fabsf(v.z) + fabsf(v.w);
    }
    mask[t * B_ + b] = (s != 0.f) ? 1.f : 0.f;
}

// ---------------------------------------------------------------------------
// XWx[t*B + b, n] = sum_k F[b,t,k] * Wx[k,n] + bias[n]
// One wave = one 16x16 tile; block = 8 waves. Uses pair-swizzled Wx.
// ---------------------------------------------------------------------------
__global__ __launch_bounds__(256) void xwx_kernel(const float* __restrict__ F,
                                                  const float* __restrict__ Wxp,
                                                  const float* __restrict__ bias,
                                                  float* __restrict__ XWx) {
    const int wave = threadIdx.x >> 5;
    const int lane = threadIdx.x & 31;
    const int lh = lane >> 4;      // K-half select for A/B frags
    const int ln = lane & 15;      // row (A) / col (B,C) within tile
    const int tile  = blockIdx.x * 8 + wave;   // [0, 65536)
    const int ntile = tile & 63;
    const int mtile = tile >> 6;               // [0, 1024)
    const int t     = mtile >> 2;              // 4 row-tiles per timestep
    const int b0    = (mtile & 3) * 16;
    const int n     = ntile * 16 + ln;

    v8f c = {0.f, 0.f, 0.f, 0.f, 0.f, 0.f, 0.f, 0.f};
    const float* arow = F + ((long)(b0 + ln) * T_ + t) * D_;   // A row b=b0+ln
    for (int k = 0; k < D_; k += 4) {
        v2f a  = *(const v2f*)(arow + k + 2 * lh);              // A[ln][k+2lh..+1]
        v2f bf = *(const v2f*)(Wxp + ((long)((k >> 1) + lh) * H_ + n) * 2);
        c = __builtin_amdgcn_wmma_f32_16x16x4_f32(false, a, false, bf,
                                                  (short)0, c, false, false);
    }
    const float bb = bias[n];
#pragma unroll
    for (int i = 0; i < 8; ++i) {
        int ml = i + 8 * lh;                                    // C row
        long g = (long)t * B_ + b0 + ml;
        XWx[g * H_ + n] = c[i] + bb;
    }
}

// ---------------------------------------------------------------------------
// Device-scope grid barrier for NBLK_ co-resident blocks.
// Monotonic counter: after step t every block has added 1 -> NBLK_*(t+1).
// ---------------------------------------------------------------------------
__device__ __forceinline__ void grid_sync(unsigned* counter, unsigned target) {
    __threadfence();                      // release: state stores visible device-wide
    __syncthreads();
    if (threadIdx.x == 0) {
        __hip_atomic_fetch_add(counter, 1u, __ATOMIC_RELEASE, __HIP_MEMORY_SCOPE_AGENT);
        while (__hip_atomic_load(counter, __ATOMIC_ACQUIRE, __HIP_MEMORY_SCOPE_AGENT)
               < target)
            __builtin_amdgcn_s_sleep(1);
    }
    __syncthreads();
    __threadfence();                      // acquire: drop stale lines before next reads
}

// ---------------------------------------------------------------------------
// Persistent scan kernel: all 256 recurrence steps in one launch.
// grid = NBLK_ blocks (16 batch rows each) x 512 threads = 16 waves; wave w
// owns columns [64w, 64w+64) as four 16x16 WMMA accumulators.
// ---------------------------------------------------------------------------
__global__ __launch_bounds__(512) void rnn_scan_kernel(
    const float* __restrict__ XWx,       // (T,B,H) pre-activations
    const float* __restrict__ S0,        // (B,H) initial state
    const float* __restrict__ Whp,       // (H/2,H,2) pair-swizzled Wh
    const float* __restrict__ Wh_halt,   // (H)
    const float* __restrict__ b_halt,    // (1)
    const float* __restrict__ maskb,     // (T,B)
    float* __restrict__ outputs,         // (T,B,H)
    float* __restrict__ states,          // (T,B,H)  (= recurrent state chain)
    unsigned* __restrict__ counter)      // grid-barrier counter (zeroed)
{
    __shared__ float lds_part[512][9];   // stride 9: conflict-free on 64 banks
    __shared__ float lds_flag[16];

    const int wave = threadIdx.x >> 5;
    const int lane = threadIdx.x & 31;
    const int lh = lane >> 4;
    const int ln = lane & 15;
    const int m0 = blockIdx.x * 16;      // batch-row base
    const int nb = wave * 64;            // column base for this wave
    const float bh0 = b_halt[0];

    for (int t = 0; t < T_; ++t) {
        const float* XWx_t    = XWx + (long)t * B_ * H_;
        const float* state_in = (t == 0) ? S0 : (states + (long)(t - 1) * B_ * H_);
        const float* mask_t   = maskb + t * B_;
        float* y_out = outputs + (long)t * B_ * H_;
        float* s_out = states  + (long)t * B_ * H_;

        // Init accumulators from XWx (C layout: row i+8*lh, col ln).
        v8f c[4];
#pragma unroll
        for (int s = 0; s < 4; ++s) {
#pragma unroll
            for (int i = 0; i < 8; ++i)
                c[s][i] = XWx_t[(long)(m0 + i + 8 * lh) * H_ + nb + s * 16 + ln];
        }

        // state @ Wh : per k-step 1 A b64 + 4 B b64 + 4 independent WMMAs.
        const float* arow = state_in + (long)(m0 + ln) * H_;
        for (int k = 0; k < H_; k += 4) {
            if ((k & 63) == 0 && k + 64 < H_)
                __builtin_prefetch(Whp + ((long)((k + 64) >> 1) * H_ + nb) * 2, 0, 1);
            v2f a = *(const v2f*)(arow + k + 2 * lh);
            const long prow = (long)((k >> 1) + lh) * H_;      // pair-row base
#pragma unroll
            for (int s = 0; s < 4; ++s) {
                const int n = nb + s * 16 + ln;
                v2f bf = *(const v2f*)(Whp + (prow + n) * 2);
                c[s] = __builtin_amdgcn_wmma_f32_16x16x4_f32(
                    false, a, false, bf, (short)0, c[s], false, false);
            }
        }

        // tanh in place + per-lane halting partial sums (8 rows per lane).
        float part[8];
#pragma unroll
        for (int i = 0; i < 8; ++i) part[i] = 0.f;
#pragma unroll
        for (int s = 0; s < 4; ++s) {
            const float wh = Wh_halt[nb + s * 16 + ln];
#pragma unroll
            for (int i = 0; i < 8; ++i) {
                float o = tanhf(c[s][i]);
                c[s][i] = o;
                part[i] += o * wh;
            }
        }
#pragma unroll
        for (int i = 0; i < 8; ++i) lds_part[threadIdx.x][i] = part[i];
        __syncthreads();

        // Deterministic fixed-order reduction: one thread per batch row r.
        if (threadIdx.x < 16) {
            const int r = threadIdx.x;
            float acc = 0.f;
            for (int w = 0; w < 16; ++w)
                for (int l = 0; l < 16; ++l)
                    acc += lds_part[w * 32 + (r >> 3) * 16 + l][r & 7];
            float p = 1.f / (1.f + expf(-(acc + bh0)));
            lds_flag[r] = (p > 0.7f) ? 1.f : 0.f;   // 1.0 - ACT_EPSILON
        }
        __syncthreads();

        // Apply flag & feature mask; write y and new_state.
#pragma unroll
        for (int s = 0; s < 4; ++s) {
            const int n = nb + s * 16 + ln;
#pragma unroll
            for (int i = 0; i < 8; ++i) {
                const int ml = i + 8 * lh;
                const float flag = lds_flag[ml];
                const float o = c[s][i] * mask_t[m0 + ml];
                const long off = (long)(m0 + ml) * H_ + n;
                y_out[off] = flag * o;
                s_out[off] = (1.f - flag) * o;
            }
        }

        // Device-wide barrier before next step consumes new_state.
        grid_sync(counter, (unsigned)(NBLK_ * (t + 1)));
    }
}

// ---------------------------------------------------------------------------
extern "C" void kernel_launch(void* const* d_in, const int* in_sizes, int n_in,
                              void* d_out, int out_size, void* d_ws, size_t ws_size,
                              hipStream_t stream) {
    const float* F    = (const float*)d_in[0];   // (B,T,D)
    const float* S0   = (const float*)d_in[1];   // (B,H)
    const float* Wx   = (const float*)d_in[2];   // (D,H)
    const float* Wh   = (const float*)d_in[3];   // (H,H)
    const float* bias = (const float*)d_in[4];   // (H)
    const float* Whh  = (const float*)d_in[5];   // (H,1)
    const float* bh   = (const float*)d_in[6];   // (1)

    float* outp = (float*)d_out;
    const long TBH = (long)T_ * B_ * H_;
    float* outputs = outp;          // (T,B,H)
    float* states  = outp + TBH;    // (T,B,H) — also the recurrent state chain

    // Workspace: XWx (64MB) | mask (64KB) | Whp (4MB) | Wxp (4MB) | counter
    float* XWx   = (float*)d_ws;
    float* maskb = XWx + TBH;
    float* Whp   = maskb + (long)T_ * B_;
    float* Wxp   = Whp + (long)H_ * H_;
    unsigned* counter = (unsigned*)(Wxp + (long)H_ * H_);

    const int swz_threads = (H_ / 2) * H_;
    swizzle_kernel<<<(swz_threads + 255) / 256, 256, 0, stream>>>(Wh, Whp);
    swizzle_kernel<<<(swz_threads + 255) / 256, 256, 0, stream>>>(Wx, Wxp);
    mask_kernel<<<(B_ * T_ + 255) / 256, 256, 0, stream>>>(F, maskb, counter);
    xwx_kernel<<<(1024 * 64) / 8, 256, 0, stream>>>(F, Wxp, bias, XWx);

    rnn_scan_kernel<<<NBLK_, 512, 0, stream>>>(XWx, S0, Whp, Whh, bh, maskb,
                                               outputs, states, counter);
}